// SAGE_NeighSampler_26749056319621
// MI455X (gfx1250) — compile-verified
//
#include <hip/hip_runtime.h>

// Problem constants (match reference)
#define NN 50000
#define NE 800000
#define DD 128
#define NC 7
#define BN_EPS 1e-5f
#define LDSW 132   // padded A/X LDS row stride (floats): even (8B v2f) + conflict-free

typedef __attribute__((ext_vector_type(2))) float v2f;
typedef __attribute__((ext_vector_type(8))) float v8f;

// ---------------------------------------------------------------------------
// In-degree counts (float, computed once, reused across all 3 layers)
// ---------------------------------------------------------------------------
__global__ void count_edges_kernel(const int* __restrict__ dst, float* __restrict__ cnt) {
  int e = blockIdx.x * blockDim.x + threadIdx.x;
  if (e < NE) unsafeAtomicAdd(&cnt[dst[e]], 1.0f);
}

// ---------------------------------------------------------------------------
// Edge scatter-add: 32 lanes per edge, one float4 per lane => one fully
// coalesced 512B global_load_b128 wavefront per edge row, then f32 atomics.
// ---------------------------------------------------------------------------
__global__ void scatter_mean_kernel(const int* __restrict__ src, const int* __restrict__ dst,
                                    const float* __restrict__ x, float* __restrict__ agg) {
  long long t = (long long)blockIdx.x * blockDim.x + threadIdx.x;
  long long e = t >> 5;
  if (e >= NE) return;
  int lane = (int)t & 31;
  long long s = src[e];
  long long d = dst[e];
  float4 v = *(const float4*)(x + s * DD + (lane << 2));
  float* o = agg + d * DD + (lane << 2);
  unsafeAtomicAdd(o + 0, v.x);
  unsafeAtomicAdd(o + 1, v.y);
  unsafeAtomicAdd(o + 2, v.z);
  unsafeAtomicAdd(o + 3, v.w);
}

// ---------------------------------------------------------------------------
// W [rows x 128] -> WT [128 x padRows], zero-padded columns (for the 7-wide head)
// ---------------------------------------------------------------------------
__global__ void transpose_pad_kernel(const float* __restrict__ W, float* __restrict__ WT,
                                     int rows, int padRows) {
  int idx = blockIdx.x * blockDim.x + threadIdx.x;
  if (idx >= DD * padRows) return;
  int k = idx / padRows;
  int n = idx - k * padRows;
  WT[idx] = (n < rows) ? W[n * DD + k] : 0.0f;
}

// ---------------------------------------------------------------------------
// Fused SAGE update GEMM:  out = act( bn( A/deg @ WaT  +  X @ WxT  + bias ) )
// Block = 128 threads (4 waves). Block owns a 16-row strip; each wave owns one
// 16x16 output tile (grid.y*64 + wave*16 columns).
//
// LDS layout (floats, dynamic):
//   [0,2112)        A strip, 16 x 128, stride 132  (scaled by 1/max(deg,1))
//   [2112,4224)     X strip (HAS_X only)
//   bOff + wave*slice : per-wave weight slices in packed fragment order:
//       packed[kk][n] = float2( W^T[2kk][col], W^T[2kk+1][col] )  (kk=0..63,n=0..15)
//   => B-fragment for K-step k0 is ONE ds_load_b64 at immediate offset 64*k0,
//      bank-conflict-free (lanes 0-15 -> banks 0-31, lanes 16-31 -> banks 32-63).
// Inner loop: 32 fully-unrolled steps of ds_load_b64 x2 + v_wmma_f32_16x16x4_f32
// per input matrix; all LDS offsets are compile-time immediates.
// ---------------------------------------------------------------------------
template <bool HAS_X>
__global__ __launch_bounds__(128) void gemm_wmma_kernel(
    const float* __restrict__ A, const float* __restrict__ cnt,
    const float* __restrict__ X,
    const float* __restrict__ WaT, const float* __restrict__ WxT, int ldBT,
    const float* __restrict__ bias, const float* __restrict__ gamma,
    const float* __restrict__ beta,
    float* __restrict__ out, int ldOut, int relu) {
  extern __shared__ float smem[];
  constexpr int A_FLOATS = 16 * LDSW;                  // 2112
  constexpr int bOff     = HAS_X ? 2 * A_FLOATS : A_FLOATS;
  constexpr int slice    = HAS_X ? 4096 : 2048;        // floats per wave (2 or 1 slices)

  const int tid  = threadIdx.x;
  const int lane = tid & 31;
  const int wave = tid >> 5;
  const long long rowBase = (long long)blockIdx.x * 16;  // NN % 16 == 0 -> always full

  // ---- Stage 16x128 strips of A (pre-scaled by 1/deg) and X, float4-coalesced.
#pragma unroll
  for (int i = 0; i < 4; ++i) {
    int linear = tid + i * 128;
    int r = linear >> 5;          // 0..15
    int c = (linear & 31) << 2;   // 0..124
    long long gr = rowBase + r;
    float4 va = *(const float4*)(A + gr * DD + c);
    if (cnt) {
      float iv = 1.0f / fmaxf(cnt[gr], 1.0f);
      va.x *= iv; va.y *= iv; va.z *= iv; va.w *= iv;
    }
    *(float4*)(&smem[r * LDSW + c]) = va;
    if constexpr (HAS_X) {
      *(float4*)(&smem[A_FLOATS + r * LDSW + c]) = *(const float4*)(X + gr * DD + c);
    }
  }

  // ---- Stage this wave's weight slice(s) in packed fragment-pair order.
  const int tileCol  = (int)blockIdx.y * 64 + wave * 16;
  const int tileColC = (tileCol < ldBT) ? tileCol : (ldBT - 16);  // clamp idle waves
  {
    float* sl = smem + bOff + wave * slice;
    const float* WA = WaT + tileColC;
#pragma unroll
    for (int j = 0; j < 8; ++j) {
      int idx = lane + j * 32;
      int kk = idx >> 2;            // 0..63 (k-pair)
      int c  = (idx & 3) << 2;      // 0,4,8,12 (column quad)
      float4 r0 = *(const float4*)(WA + (2 * kk) * ldBT + c);
      float4 r1 = *(const float4*)(WA + (2 * kk + 1) * ldBT + c);
      float4 p0 = {r0.x, r1.x, r0.y, r1.y};
      float4 p1 = {r0.z, r1.z, r0.w, r1.w};
      float* dp = sl + kk * 32 + c * 2;
      *(float4*)(dp)     = p0;
      *(float4*)(dp + 4) = p1;
    }
    if constexpr (HAS_X) {
      float* slx = sl + 2048;
      const float* WX = WxT + tileColC;
#pragma unroll
      for (int j = 0; j < 8; ++j) {
        int idx = lane + j * 32;
        int kk = idx >> 2;
        int c  = (idx & 3) << 2;
        float4 r0 = *(const float4*)(WX + (2 * kk) * ldBT + c);
        float4 r1 = *(const float4*)(WX + (2 * kk + 1) * ldBT + c);
        float4 p0 = {r0.x, r1.x, r0.y, r1.y};
        float4 p1 = {r0.z, r1.z, r0.w, r1.w};
        float* dp = slx + kk * 32 + c * 2;
        *(float4*)(dp)     = p0;
        *(float4*)(dp + 4) = p1;
      }
    }
  }
  __syncthreads();

  if (tileCol < ldBT) {  // wave-uniform: EXEC stays all-ones around WMMA
    const int m = lane & 15;              // A row within tile
    const int h = lane >> 4;              // K half: frag K-offset 2h
    const int n = lane & 15;              // B/D column within tile

    v8f acc = {0.f, 0.f, 0.f, 0.f, 0.f, 0.f, 0.f, 0.f};
    const float* aBase = smem + m * LDSW + 2 * h;                      // + 4*s
    const float* bBase = smem + bOff + wave * slice + 32 * h + 2 * n;  // + 16*s

#pragma unroll
    for (int s = 0; s < 32; ++s) {           // K = 4*s
      v2f a = *(const v2f*)(aBase + 4 * s);             // ds_load_b64, imm offset
      v2f b = *(const v2f*)(bBase + 16 * s);            // ds_load_b64, imm offset
      acc = __builtin_amdgcn_wmma_f32_16x16x4_f32(false, a, false, b,
                                                  (short)0, acc, false, false);
      if constexpr (HAS_X) {
        v2f ax = *(const v2f*)(aBase + A_FLOATS + 4 * s);
        v2f bx = *(const v2f*)(bBase + 2048 + 16 * s);
        acc = __builtin_amdgcn_wmma_f32_16x16x4_f32(false, ax, false, bx,
                                                    (short)0, acc, false, false);
      }
    }

    // Epilogue: fold bias + eval BN (y*g/sqrt(1+eps) + beta) + optional ReLU.
    const int c = tileCol + n;
    if (c < ldOut) {
      float scale = 1.0f;
      if (gamma) scale = gamma[c] * rsqrtf(1.0f + BN_EPS);
      float shift = (bias ? bias[c] : 0.0f) * scale + (beta ? beta[c] : 0.0f);
      const int rbase = h << 3;  // D-matrix: VGPR v -> row v + 8*(lane>=16)
#pragma unroll
      for (int v = 0; v < 8; ++v) {
        float r = acc[v] * scale + shift;
        if (relu) r = fmaxf(r, 0.0f);
        out[(rowBase + rbase + v) * (long long)ldOut + c] = r;
      }
    }
  }
}

// ---------------------------------------------------------------------------
extern "C" void kernel_launch(void* const* d_in, const int* in_sizes, int n_in,
                              void* d_out, int out_size, void* d_ws, size_t ws_size,
                              hipStream_t stream) {
  const float* x   = (const float*)d_in[0];
  const int*  eix  = (const int*)d_in[1];
  const float *Wl0 = (const float*)d_in[2],  *Wr0 = (const float*)d_in[3],  *b0 = (const float*)d_in[4];
  const float *Wl1 = (const float*)d_in[5],  *Wr1 = (const float*)d_in[6],  *b1 = (const float*)d_in[7];
  const float *Wl2 = (const float*)d_in[8],  *Wr2 = (const float*)d_in[9],  *b2 = (const float*)d_in[10];
  const float *g0  = (const float*)d_in[11], *be0 = (const float*)d_in[12];
  const float *g1  = (const float*)d_in[13], *be1 = (const float*)d_in[14];
  const float *dW1 = (const float*)d_in[15], *db1 = (const float*)d_in[16];
  const float *dW2 = (const float*)d_in[17], *db2 = (const float*)d_in[18];
  const float *dW3 = (const float*)d_in[19], *db3 = (const float*)d_in[20];
  const int* srcI = eix;        // edge_index[0]
  const int* dstI = eix + NE;   // edge_index[1]
  float* out = (float*)d_out;

  // Workspace carve-up (256B aligned): ~77 MB total
  char* ws = (char*)d_ws;
  size_t off = 0;
  auto take = [&](size_t bytes) -> float* {
    float* p = (float*)(ws + off);
    off = (off + bytes + 255) & ~(size_t)255;
    return p;
  };
  float* cnt  = take((size_t)NN * 4);
  float* bufA = take((size_t)NN * DD * 4);  // neighbor sums (reused each layer)
  float* bufB = take((size_t)NN * DD * 4);  // h0, later h2
  float* bufC = take((size_t)NN * DD * 4);  // h1
  float* WlT0 = take(DD * DD * 4);  float* WrT0 = take(DD * DD * 4);
  float* WlT1 = take(DD * DD * 4);  float* WrT1 = take(DD * DD * 4);
  float* WlT2 = take(DD * DD * 4);  float* WrT2 = take(DD * DD * 4);
  float* dW1T = take(DD * DD * 4);  float* dW3T = take(DD * DD * 4);
  float* dW2T = take(DD * 16 * 4);  // 7 -> padded to 16 cols

  // Degree counts: computed once, reused by all three layers.
  hipMemsetAsync(cnt, 0, (size_t)NN * 4, stream);
  count_edges_kernel<<<(NE + 255) / 256, 256, 0, stream>>>(dstI, cnt);

  // Pre-transpose all weights so staging loads are row-contiguous float4s.
  const int tBlk = (DD * DD + 255) / 256;
  transpose_pad_kernel<<<tBlk, 256, 0, stream>>>(Wl0, WlT0, DD, DD);
  transpose_pad_kernel<<<tBlk, 256, 0, stream>>>(Wr0, WrT0, DD, DD);
  transpose_pad_kernel<<<tBlk, 256, 0, stream>>>(Wl1, WlT1, DD, DD);
  transpose_pad_kernel<<<tBlk, 256, 0, stream>>>(Wr1, WrT1, DD, DD);
  transpose_pad_kernel<<<tBlk, 256, 0, stream>>>(Wl2, WlT2, DD, DD);
  transpose_pad_kernel<<<tBlk, 256, 0, stream>>>(Wr2, WrT2, DD, DD);
  transpose_pad_kernel<<<tBlk, 256, 0, stream>>>(dW1, dW1T, DD, DD);
  transpose_pad_kernel<<<tBlk, 256, 0, stream>>>(dW3, dW3T, DD, DD);
  transpose_pad_kernel<<<(DD * 16 + 255) / 256, 256, 0, stream>>>(dW2, dW2T, NC, 16);

  const unsigned scBlocks = (unsigned)(((long long)NE * 32 + 255) / 256);
  const dim3 gemmGrid2(NN / 16, 2);  // 8 column tiles / 4 waves
  const dim3 gemmGrid1(NN / 16, 1);  // padded 16-wide head
  const size_t smX  = (size_t)(2 * 16 * LDSW + 4 * 4096) * 4;  // 82,432 B
  const size_t smNX = (size_t)(16 * LDSW + 4 * 2048) * 4;      // 41,216 B

  // ---- Layer 0: h0 = relu(bn0(mean_agg(x) @ Wl0^T + x @ Wr0^T + b0))
  hipMemsetAsync(bufA, 0, (size_t)NN * DD * 4, stream);
  scatter_mean_kernel<<<scBlocks, 256, 0, stream>>>(srcI, dstI, x, bufA);
  gemm_wmma_kernel<true><<<gemmGrid2, 128, smX, stream>>>(bufA, cnt, x, WlT0, WrT0, DD,
                                                          b0, g0, be0, bufB, DD, 1);
  // ---- Layer 1: h1 = relu(bn1(mean_agg(h0) @ Wl1^T + h0 @ Wr1^T + b1))
  hipMemsetAsync(bufA, 0, (size_t)NN * DD * 4, stream);
  scatter_mean_kernel<<<scBlocks, 256, 0, stream>>>(srcI, dstI, bufB, bufA);
  gemm_wmma_kernel<true><<<gemmGrid2, 128, smX, stream>>>(bufA, cnt, bufB, WlT1, WrT1, DD,
                                                          b1, g1, be1, bufC, DD, 1);
  // ---- Layer 2: h2 = mean_agg(h1) @ Wl2^T + h1 @ Wr2^T + b2   (no BN/ReLU)
  hipMemsetAsync(bufA, 0, (size_t)NN * DD * 4, stream);
  scatter_mean_kernel<<<scBlocks, 256, 0, stream>>>(srcI, dstI, bufC, bufA);
  gemm_wmma_kernel<true><<<gemmGrid2, 128, smX, stream>>>(bufA, cnt, bufC, WlT2, WrT2, DD,
                                                          b2, nullptr, nullptr, bufB, DD, 0);

  // ---- Heads (outputs concatenated flat in return order)
  float* out1 = out;                               // [NN x 128]
  float* out2 = out + (size_t)NN * DD;             // [NN x 7]
  float* out3 = out + (size_t)NN * (DD + NC);      // [NN x 128]
  gemm_wmma_kernel<false><<<gemmGrid2, 128, smNX, stream>>>(bufB, nullptr, nullptr, dW1T, nullptr, DD,
                                                            db1, nullptr, nullptr, out1, DD, 0);
  gemm_wmma_kernel<false><<<gemmGrid1, 128, smNX, stream>>>(bufB, nullptr, nullptr, dW2T, nullptr, 16,
                                                            db2, nullptr, nullptr, out2, NC, 0);
  gemm_wmma_kernel<false><<<gemmGrid2, 128, smNX, stream>>>(bufB, nullptr, nullptr, dW3T, nullptr, DD,
                                                            db3, nullptr, nullptr, out3, DD, 0);
}